// MHANet_44092134260997
// MI455X (gfx1250) — compile-verified
//
#include <hip/hip_runtime.h>

typedef float v2f __attribute__((ext_vector_type(2)));
typedef float v8f __attribute__((ext_vector_type(8)));

static constexpr int BATCH = 16;
static constexpr int DIMC  = 256;
static constexpr int SEQ   = 4096;
static constexpr int HEADS = 16;
static constexpr int QKVC  = 3 * DIMC;   // 768

#define DEV __device__ __forceinline__

// ---------------------------------------------------------------------------
// block reduction helper (blockDim.x must be a power of two, shared float[256])
// ---------------------------------------------------------------------------
DEV float block_reduce_sum(float v, float* sh) {
    int tid = threadIdx.x;
    sh[tid] = v;
    __syncthreads();
    for (int s = blockDim.x >> 1; s > 0; s >>= 1) {
        if (tid < s) sh[tid] += sh[tid + s];
        __syncthreads();
    }
    float r = sh[0];
    __syncthreads();
    return r;
}

DEV float eluf(float x)   { return x > 0.f ? x : expm1f(x); }
DEV float lrelu(float x)  { return x > 0.f ? x : 0.01f * x; }
DEV int   clampi(int v, int lo, int hi) { return v < lo ? lo : (v > hi ? hi : v); }

// ---------------------------------------------------------------------------
// Generic batched WMMA f32 GEMM:  C[b] = A[b](MxK) * B[b](KxN) (+bias, B rowscale)
// One wave computes MB vertical 16x16 tiles of one 16-wide N tile.
// M, N, K multiples of 16 / 4 for all uses here.
// ---------------------------------------------------------------------------
template <int MB>
__global__ void gemm_wmma_f32(const float* __restrict__ A, int lda, long sA,
                              const float* __restrict__ B, int ldb, long sB,
                              float* __restrict__ C, int ldc, long sC,
                              const float* __restrict__ bias,
                              const float* __restrict__ rowscaleB,  // stride K per batch
                              int M, int N, int K) {
    const int wave  = threadIdx.x >> 5;
    const int lane  = threadIdx.x & 31;
    const int ntile = blockIdx.x * (blockDim.x >> 5) + wave;
    if (ntile * 16 >= N) return;                 // wave-uniform
    const int mbase = blockIdx.y * (16 * MB);
    const int batch = blockIdx.z;
    const int r     = lane & 15;
    const int half  = lane >> 4;
    const int col   = ntile * 16 + r;

    const float* Ab = A + (size_t)batch * sA;
    const float* Bb = B + (size_t)batch * sB;
    float*       Cb = C + (size_t)batch * sC;
    const float* rs = rowscaleB ? rowscaleB + (size_t)batch * K : nullptr;

    v8f acc[MB];
#pragma unroll
    for (int i = 0; i < MB; ++i)
#pragma unroll
        for (int e = 0; e < 8; ++e) acc[i][e] = 0.f;

    for (int k0 = 0; k0 < K; k0 += 4) {
        const int ka = k0 + 2 * half;
        v2f bv;
        bv.x = Bb[(size_t)ka * ldb + col];
        bv.y = Bb[(size_t)(ka + 1) * ldb + col];
        if (rs) { bv.x *= rs[ka]; bv.y *= rs[ka + 1]; }
#pragma unroll
        for (int i = 0; i < MB; ++i) {
            const int row = mbase + i * 16 + r;
            v2f av;
            av.x = Ab[(size_t)row * lda + ka];
            av.y = Ab[(size_t)row * lda + ka + 1];
            acc[i] = __builtin_amdgcn_wmma_f32_16x16x4_f32(
                false, av, false, bv, (short)0, acc[i], false, false);
        }
    }
#pragma unroll
    for (int i = 0; i < MB; ++i)
#pragma unroll
        for (int e = 0; e < 8; ++e) {
            const int row = mbase + i * 16 + e + 8 * half;
            float v = acc[i][e];
            if (bias) v += bias[row];
            Cb[(size_t)row * ldc + col] = v;
        }
}

// ---------------------------------------------------------------------------
// depthwise 3x3 conv on (B,768,SEQ,1): only center column of the 3x3 matters.
// clamped loads + multiplicative masks (no divergent branches around loads)
// ---------------------------------------------------------------------------
__global__ void dwconv_kernel(const float* __restrict__ in,
                              const float* __restrict__ w,
                              float* __restrict__ out) {
    size_t idx = (size_t)blockIdx.x * blockDim.x + threadIdx.x;
    if (idx >= (size_t)BATCH * QKVC * SEQ) return;
    int s = (int)(idx % SEQ);
    int o = (int)((idx / SEQ) % QKVC);
    float w0 = w[o * 9 + 1], w1 = w[o * 9 + 4], w2 = w[o * 9 + 7];
    float mm = (s > 0) ? 1.f : 0.f;
    float mp = (s < SEQ - 1) ? 1.f : 0.f;
    float xm = in[idx - (s > 0 ? 1 : 0)];
    float xc = in[idx];
    float xp = in[idx + (s < SEQ - 1 ? 1 : 0)];
    out[idx] = w0 * mm * xm + w1 * xc + w2 * mp * xp;
}

// ---------------------------------------------------------------------------
// per-row inverse L2 norms of q,k (rows of length SEQ). grid.x = 2*B*DIM blocks
// ---------------------------------------------------------------------------
__global__ void rownorm_kernel(const float* __restrict__ qkv,
                               float* __restrict__ invq, float* __restrict__ invk) {
    __shared__ float sh[256];
    int row = blockIdx.x;
    int which = row >= BATCH * DIMC;
    int rr = row - which * BATCH * DIMC;
    int b = rr / DIMC, c = rr % DIMC;
    const float* p = qkv + ((size_t)b * QKVC + which * DIMC + c) * SEQ;
    float s = 0.f;
    for (int i = threadIdx.x; i < SEQ; i += blockDim.x) { float v = p[i]; s += v * v; }
    s = block_reduce_sum(s, sh);
    if (threadIdx.x == 0) {
        float n = fmaxf(sqrtf(s), 1e-12f);
        (which ? invk : invq)[b * DIMC + c] = 1.f / n;
    }
}

// ---------------------------------------------------------------------------
// per-position mean over each band-select chunk's channels of v
// mu3 layout: (chunk, b, s)
// ---------------------------------------------------------------------------
__global__ void chunk_mean_kernel(const float* __restrict__ qkv, float* __restrict__ mu3) {
    int idx = blockIdx.x * blockDim.x + threadIdx.x;
    if (idx >= 3 * BATCH * SEQ) return;
    int s = idx % SEQ;
    int cb = idx / SEQ;
    int b = cb % BATCH;
    int chunk = cb / BATCH;
    int c0 = (chunk == 0) ? 0 : (chunk == 1 ? 86 : 171);
    int Ci = (chunk == 0) ? 86 : 85;
    const float* p = qkv + ((size_t)b * QKVC + 2 * DIMC + c0) * SEQ + s;
    float acc = 0.f;
    for (int c = 0; c < Ci; ++c) acc += p[(size_t)c * SEQ];
    mu3[idx] = acc / (float)Ci;
}

// ---------------------------------------------------------------------------
// covariance Gram via WMMA f32: cov[b,chunk] = Xm * Xm^T, Xm = (v - mu)/4095
// 86/85 channels padded to 96 (6x6 tiles of 16). covbuf: (b*3+chunk, 96, 96)
// invalid (padded) rows handled by clamped pointer + loop-invariant 0/inv mask,
// so all loads are unconditional and the K loop has no exec-mask churn.
// ---------------------------------------------------------------------------
__global__ void gram_wmma_kernel(const float* __restrict__ qkv,
                                 const float* __restrict__ mu3,
                                 float* __restrict__ covbuf) {
    const int wave = threadIdx.x >> 5, lane = threadIdx.x & 31;
    const int tile = blockIdx.x * (blockDim.x >> 5) + wave;
    if (tile >= 36) return;
    const int ti = tile / 6, tj = tile % 6;
    const int bc = blockIdx.y;
    const int b = bc / 3, chunk = bc % 3;
    const int c0 = (chunk == 0) ? 0 : (chunk == 1 ? 86 : 171);
    const int Ci = (chunk == 0) ? 86 : 85;
    const int r = lane & 15, half = lane >> 4;
    const int ri = ti * 16 + r, rj = tj * 16 + r;
    const float inv = 1.f / 4095.f;
    const float mA = (ri < Ci) ? inv : 0.f;     // loop-invariant masks
    const float mB = (rj < Ci) ? inv : 0.f;
    const int rci = (ri < Ci) ? ri : 0;
    const int rcj = (rj < Ci) ? rj : 0;
    const float* vbase = qkv + ((size_t)b * QKVC + 2 * DIMC + c0) * SEQ;
    const float* pi = vbase + (size_t)rci * SEQ;
    const float* pj = vbase + (size_t)rcj * SEQ;
    const float* mub = mu3 + ((size_t)chunk * BATCH + b) * SEQ;

    v8f acc;
#pragma unroll
    for (int e = 0; e < 8; ++e) acc[e] = 0.f;

#pragma unroll 2
    for (int k0 = 0; k0 < SEQ; k0 += 4) {
        const int ka = k0 + 2 * half;
        float mu0 = mub[ka], mu1 = mub[ka + 1];
        v2f a, bv;
        a.x  = (pi[ka] - mu0) * mA;
        a.y  = (pi[ka + 1] - mu1) * mA;
        bv.x = (pj[ka] - mu0) * mB;
        bv.y = (pj[ka + 1] - mu1) * mB;
        acc = __builtin_amdgcn_wmma_f32_16x16x4_f32(
            false, a, false, bv, (short)0, acc, false, false);
    }
    float* cb = covbuf + (size_t)bc * 96 * 96;
#pragma unroll
    for (int e = 0; e < 8; ++e)
        cb[(size_t)(ti * 16 + e + 8 * half) * 96 + tj * 16 + r] = acc[e];
}

// ---------------------------------------------------------------------------
// cov normalization, stat, 3-layer MLP -> channel weights. grid 48, blockDim 96
// ---------------------------------------------------------------------------
struct CovParams {
    const float* w1[3]; const float* b1[3];
    const float* w2[3]; const float* b2[3];
    const float* w3[3]; const float* b3[3];
};

__global__ void cov_stat_mlp_kernel(const float* __restrict__ covbuf,
                                    CovParams cp, float* __restrict__ wgt) {
    __shared__ float nrm[96], st[96], h1s[96], h2s[64];
    const int bc = blockIdx.x;
    const int b = bc / 3, chunk = bc % 3;
    const int c0 = (chunk == 0) ? 0 : (chunk == 1 ? 86 : 171);
    const int Ci = (chunk == 0) ? 86 : 85;
    const int HD = 51;
    const float* cov = covbuf + (size_t)bc * 96 * 96;
    const int c = threadIdx.x;
    if (c < Ci) nrm[c] = sqrtf(cov[(size_t)c * 96 + c]);
    __syncthreads();
    if (c < Ci) {
        float s = 0.f;
        for (int d = 0; d < Ci; ++d)
            s += cov[(size_t)c * 96 + d] / (nrm[c] * nrm[d] + 1e-6f);
        st[c] = s / (float)Ci;
    }
    __syncthreads();
    if (c < Ci) {
        float t = cp.b1[chunk][c];
        const float* w = cp.w1[chunk] + (size_t)c * Ci;
        for (int d = 0; d < Ci; ++d) t += w[d] * st[d];
        h1s[c] = lrelu(t);
    }
    __syncthreads();
    if (c < HD) {
        float t = cp.b2[chunk][c];
        const float* w = cp.w2[chunk] + (size_t)c * Ci;
        for (int d = 0; d < Ci; ++d) t += w[d] * h1s[d];
        h2s[c] = lrelu(t);
    }
    __syncthreads();
    if (c < Ci) {
        float t = cp.b3[chunk][c];
        const float* w = cp.w3[chunk] + (size_t)c * HD;
        for (int j = 0; j < HD; ++j) t += w[j] * h2s[j];
        wgt[b * DIMC + c0 + c] = t;
    }
}

// ---------------------------------------------------------------------------
// spatio 1x1 conv 256->1:  xs[b,s] = sum_c w[c]*vb[b,c,s] + b0
// ---------------------------------------------------------------------------
__global__ void spatio_kernel(const float* __restrict__ vb,
                              const float* __restrict__ sw, const float* __restrict__ sb,
                              float* __restrict__ xs) {
    int idx = blockIdx.x * blockDim.x + threadIdx.x;
    if (idx >= BATCH * SEQ) return;
    int s = idx % SEQ, b = idx / SEQ;
    const float* p = vb + (size_t)b * DIMC * SEQ + s;
    float acc = sb[0];
#pragma unroll 4
    for (int c = 0; c < DIMC; ++c) acc += sw[c] * p[(size_t)c * SEQ];
    xs[idx] = acc;
}

// ---------------------------------------------------------------------------
// one MTA scale: conv1d(ksz) + LayerNorm(SEQ) + ELU + avgpool -> scalar a[b,j]
// grid (B, 3), blockDim 256, 16 positions per thread
// ---------------------------------------------------------------------------
struct MtaParams {
    const float* w[3]; const float* b[3]; const float* g[3]; const float* be[3];
};

__global__ void mta_scale_kernel(const float* __restrict__ xs,
                                 const float* __restrict__ up_w,
                                 const float* __restrict__ up_b,
                                 MtaParams mp, float* __restrict__ ascale) {
    __shared__ float sh[256];
    const int b = blockIdx.x, j = blockIdx.y;
    const int tid = threadIdx.x;
    const int ksz = 2 * (j + 1);
    const int padL = j;                         // {2:(0,1), 4:(1,2), 6:(2,3)}
    const float* w = mp.w[j];
    const float cb = mp.b[j][0];
    const float uw = up_w[j], ub = up_b[j];
    const float* xb = xs + (size_t)b * SEQ;

    float tv[16];
    float s1 = 0.f, s2 = 0.f;
    for (int ii = 0; ii < 16; ++ii) {
        int s = ii * 256 + tid;
        float t = cb;
        for (int k = 0; k < ksz; ++k) {
            int sp = s + k - padL;
            float msk = (sp >= 0 && sp < SEQ) ? 1.f : 0.f;
            int spc = clampi(sp, 0, SEQ - 1);
            t += w[k] * msk * (uw * xb[spc] + ub);
        }
        tv[ii] = t;
        s1 += t;
        s2 += t * t;
    }
    float tot  = block_reduce_sum(s1, sh);
    float tot2 = block_reduce_sum(s2, sh);
    float mu = tot / (float)SEQ;
    float var = tot2 / (float)SEQ - mu * mu;
    float rstd = rsqrtf(var + 1e-5f);
    const float* g = mp.g[j];
    const float* be = mp.be[j];
    float se = 0.f;
    for (int ii = 0; ii < 16; ++ii) {
        int s = ii * 256 + tid;
        float tn = (tv[ii] - mu) * rstd * g[s] + be[s];
        se += eluf(tn);
    }
    se = block_reduce_sum(se, sh);
    if (tid == 0) ascale[b * 3 + j] = se / (float)SEQ;
}

// ---------------------------------------------------------------------------
// v2 = (mta_proj_w[c]*sum_j a_j*(uw_j*xs+ub_j) + mta_proj_b[c]) * vb   (in place)
// ---------------------------------------------------------------------------
__global__ void mta_apply_kernel(float* __restrict__ vb,
                                 const float* __restrict__ xs,
                                 const float* __restrict__ ascale,
                                 const float* __restrict__ up_w,
                                 const float* __restrict__ up_b,
                                 const float* __restrict__ pw,
                                 const float* __restrict__ pb) {
    size_t idx = (size_t)blockIdx.x * blockDim.x + threadIdx.x;
    if (idx >= (size_t)BATCH * DIMC * SEQ) return;
    int s = (int)(idx % SEQ);
    size_t bc = idx / SEQ;
    int c = (int)(bc % DIMC);
    int b = (int)(bc / DIMC);
    float xsv = xs[b * SEQ + s];
    float m = 0.f;
#pragma unroll
    for (int j = 0; j < 3; ++j)
        m += ascale[b * 3 + j] * (up_w[j] * xsv + up_b[j]);
    vb[idx] = (pw[c] * m + pb[c]) * vb[idx];
}

// ---------------------------------------------------------------------------
// attention logits via WMMA: logits[bh,c,d] = (q_c/|q_c|) . (k_d/|k_d|)
// one wave per (b,h). grid.x=32, blockDim 256
// ---------------------------------------------------------------------------
__global__ void attn_qk_kernel(const float* __restrict__ qkv,
                               const float* __restrict__ invq,
                               const float* __restrict__ invk,
                               float* __restrict__ logits) {
    const int wave = threadIdx.x >> 5, lane = threadIdx.x & 31;
    const int bh = blockIdx.x * (blockDim.x >> 5) + wave;
    if (bh >= BATCH * HEADS) return;
    const int b = bh >> 4, h = bh & 15;
    const int r = lane & 15, half = lane >> 4;
    const float* qrow = qkv + ((size_t)b * QKVC + h * 16 + r) * SEQ;
    const float* krow = qkv + ((size_t)b * QKVC + DIMC + h * 16 + r) * SEQ;
    const float iq = invq[b * DIMC + h * 16 + r];
    const float ik = invk[b * DIMC + h * 16 + r];

    v8f acc;
#pragma unroll
    for (int e = 0; e < 8; ++e) acc[e] = 0.f;

#pragma unroll 2
    for (int k0 = 0; k0 < SEQ; k0 += 4) {
        const int ka = k0 + 2 * half;
        v2f a, bv;
        a.x  = qrow[ka] * iq;     a.y  = qrow[ka + 1] * iq;
        bv.x = krow[ka] * ik;     bv.y = krow[ka + 1] * ik;
        acc = __builtin_amdgcn_wmma_f32_16x16x4_f32(
            false, a, false, bv, (short)0, acc, false, false);
    }
    float* lp = logits + (size_t)bh * 256;
#pragma unroll
    for (int e = 0; e < 8; ++e)
        lp[(e + 8 * half) * 16 + r] = acc[e];
}

// ---------------------------------------------------------------------------
// softmax over d (16) with temperature[h], in place. B*H*16 rows.
// ---------------------------------------------------------------------------
__global__ void softmax_kernel(float* __restrict__ logits,
                               const float* __restrict__ temperature) {
    int g = blockIdx.x * blockDim.x + threadIdx.x;
    if (g >= BATCH * HEADS * 16) return;
    int bh = g >> 4;
    int h = bh & 15;
    float t = temperature[h];
    float* row = logits + (size_t)bh * 256 + (g & 15) * 16;
    float v[16], m = -3.4e38f;
#pragma unroll
    for (int d = 0; d < 16; ++d) { v[d] = row[d] * t; m = fmaxf(m, v[d]); }
    float s = 0.f;
#pragma unroll
    for (int d = 0; d < 16; ++d) { v[d] = expf(v[d] - m); s += v[d]; }
    float inv = 1.f / s;
#pragma unroll
    for (int d = 0; d < 16; ++d) row[d] = v[d] * inv;
}

// ---------------------------------------------------------------------------
// fused MGA: BN -> up(1->3) -> {3x3, 5x5 d2, 7x7 d3} gated attn -> down + skip
// convs collapsed: a_j = w_j*S_j + b_j*W_j + cb_j with masked tap sums S_j,W_j.
// clamped addressing + 0/1 masks -> unconditional loads, no exec-mask blocks.
// ---------------------------------------------------------------------------
__global__ void mga_kernel(const float* __restrict__ y, float* __restrict__ out,
                           const float* bn_g, const float* bn_b,
                           const float* up_w, const float* up_b,
                           const float* c0w, const float* c0b,
                           const float* c1w, const float* c1b,
                           const float* c2w, const float* c2b,
                           const float* dnw, const float* dnb) {
    size_t idx = (size_t)blockIdx.x * blockDim.x + threadIdx.x;
    if (idx >= (size_t)BATCH * DIMC * SEQ) return;
    const int ww = (int)(idx % SEQ);
    size_t t = idx / SEQ;
    const int hh = (int)(t % DIMC);
    const int b = (int)(t / DIMC);
    const float* img = y + (size_t)b * DIMC * SEQ;
    const float bsc = bn_g[0] * rsqrtf(1.f + 1e-5f);
    const float bbs = bn_b[0];

    float S[3], Wm[3];
#pragma unroll
    for (int j = 0; j < 3; ++j) { S[j] = 0.f; Wm[j] = 0.f; }

    // 3x3, pad 1
#pragma unroll
    for (int kh = 0; kh < 3; ++kh)
#pragma unroll
        for (int kw = 0; kw < 3; ++kw) {
            float cw = c0w[kh * 3 + kw];
            int h2 = hh + kh - 1, w2 = ww + kw - 1;
            float msk = (h2 >= 0 && h2 < DIMC && w2 >= 0 && w2 < SEQ) ? 1.f : 0.f;
            int hc = clampi(h2, 0, DIMC - 1), wc = clampi(w2, 0, SEQ - 1);
            S[0] += cw * msk * (img[(size_t)hc * SEQ + wc] * bsc + bbs);
            Wm[0] += cw * msk;
        }
    // 5x5, dilation 2, pad 4
#pragma unroll
    for (int kh = 0; kh < 5; ++kh)
#pragma unroll
        for (int kw = 0; kw < 5; ++kw) {
            float cw = c1w[kh * 5 + kw];
            int h2 = hh + (kh - 2) * 2, w2 = ww + (kw - 2) * 2;
            float msk = (h2 >= 0 && h2 < DIMC && w2 >= 0 && w2 < SEQ) ? 1.f : 0.f;
            int hc = clampi(h2, 0, DIMC - 1), wc = clampi(w2, 0, SEQ - 1);
            S[1] += cw * msk * (img[(size_t)hc * SEQ + wc] * bsc + bbs);
            Wm[1] += cw * msk;
        }
    // 7x7, dilation 3, pad 9
#pragma unroll
    for (int kh = 0; kh < 7; ++kh)
#pragma unroll
        for (int kw = 0; kw < 7; ++kw) {
            float cw = c2w[kh * 7 + kw];
            int h2 = hh + (kh - 3) * 3, w2 = ww + (kw - 3) * 3;
            float msk = (h2 >= 0 && h2 < DIMC && w2 >= 0 && w2 < SEQ) ? 1.f : 0.f;
            int hc = clampi(h2, 0, DIMC - 1), wc = clampi(w2, 0, SEQ - 1);
            S[2] += cw * msk * (img[(size_t)hc * SEQ + wc] * bsc + bbs);
            Wm[2] += cw * msk;
        }

    const float xc = img[(size_t)hh * SEQ + ww];
    const float xbc = xc * bsc + bbs;
    float d = dnb[0] + xc;   // down bias + shortcut
    const float cbv[3] = {c0b[0], c1b[0], c2b[0]};
#pragma unroll
    for (int j = 0; j < 3; ++j) {
        float yj = up_w[j] * xbc + up_b[j];
        float aj = up_w[j] * S[j] + up_b[j] * Wm[j] + cbv[j];
        d += dnw[j] * (yj * aj);
    }
    out[idx] = d;
}

// ---------------------------------------------------------------------------
// stconv fused: temporal conv (1x2)+BN+ELU -> spatial conv (256x1)+BN+ELU
// grid (ceil(4095/256), B), s_w staged in LDS (25.6KB)
// ---------------------------------------------------------------------------
__global__ void stconv_kernel(const float* __restrict__ m,
                              const float* t_w, const float* t_b,
                              const float* bn1_g, const float* bn1_b,
                              const float* s_w, const float* s_b,
                              const float* bn2_g, const float* bn2_b,
                              float* __restrict__ out2) {
    __shared__ float sw[5 * 5 * DIMC];
    for (int i = threadIdx.x; i < 5 * 5 * DIMC; i += blockDim.x) sw[i] = s_w[i];
    __syncthreads();
    const int b = blockIdx.y;
    const int ww = blockIdx.x * blockDim.x + threadIdx.x;
    if (ww >= SEQ - 1) return;
    const float bnr = rsqrtf(1.f + 1e-5f);
    float tw0[5], tw1[5], tb[5], g1[5], b1[5];
#pragma unroll
    for (int i = 0; i < 5; ++i) {
        tw0[i] = t_w[i * 2];
        tw1[i] = t_w[i * 2 + 1];
        tb[i]  = t_b[i];
        g1[i]  = bn1_g[i] * bnr;
        b1[i]  = bn1_b[i];
    }
    float acc[5] = {0.f, 0.f, 0.f, 0.f, 0.f};
    const float* mb = m + (size_t)b * DIMC * SEQ;
    for (int kh = 0; kh < DIMC; ++kh) {
        float m0 = mb[(size_t)kh * SEQ + ww];
        float m1 = mb[(size_t)kh * SEQ + ww + 1];
#pragma unroll
        for (int i = 0; i < 5; ++i) {
            float tv = tw0[i] * m0 + tw1[i] * m1 + tb[i];
            tv = eluf(tv * g1[i] + b1[i]);
#pragma unroll
            for (int o = 0; o < 5; ++o)
                acc[o] += sw[(o * 5 + i) * DIMC + kh] * tv;
        }
    }
#pragma unroll
    for (int o = 0; o < 5; ++o) {
        float v = acc[o] + s_b[o];
        v = eluf(v * (bn2_g[o] * bnr) + bn2_b[o]);
        out2[((size_t)b * 5 + o) * (SEQ - 1) + ww] = v;
    }
}

// ---------------------------------------------------------------------------
// adaptive avg pool (1x4095) + final linear 5->2
// ---------------------------------------------------------------------------
__global__ void final_kernel(const float* __restrict__ out2,
                             const float* out_w, const float* out_b,
                             float* __restrict__ dout) {
    __shared__ float sh[256];
    const int b = blockIdx.x;
    float pool[5];
    for (int o = 0; o < 5; ++o) {
        const float* p = out2 + ((size_t)b * 5 + o) * (SEQ - 1);
        float s = 0.f;
        for (int i = threadIdx.x; i < SEQ - 1; i += blockDim.x) s += p[i];
        s = block_reduce_sum(s, sh);
        pool[o] = s / (float)(SEQ - 1);
    }
    if (threadIdx.x == 0) {
        for (int j = 0; j < 2; ++j) {
            float v = out_b[j];
            for (int i = 0; i < 5; ++i) v += out_w[j * 5 + i] * pool[i];
            dout[b * 2 + j] = v;
        }
    }
}

// ---------------------------------------------------------------------------
// host launcher
// ---------------------------------------------------------------------------
extern "C" void kernel_launch(void* const* d_in, const int* in_sizes, int n_in,
                              void* d_out, int out_size, void* d_ws, size_t ws_size,
                              hipStream_t stream) {
    (void)in_sizes; (void)n_in; (void)out_size; (void)ws_size;
    const float* x           = (const float*)d_in[0];
    const float* qkv_w       = (const float*)d_in[1];
    const float* qkv_dw_w    = (const float*)d_in[2];
    const float* temperature = (const float*)d_in[3];
    const float* proj_w      = (const float*)d_in[4];
    CovParams cp;
    for (int i = 0; i < 3; ++i) {
        cp.w1[i] = (const float*)d_in[5 + 6 * i];
        cp.b1[i] = (const float*)d_in[6 + 6 * i];
        cp.w2[i] = (const float*)d_in[7 + 6 * i];
        cp.b2[i] = (const float*)d_in[8 + 6 * i];
        cp.w3[i] = (const float*)d_in[9 + 6 * i];
        cp.b3[i] = (const float*)d_in[10 + 6 * i];
    }
    const float* chproj_w   = (const float*)d_in[23];
    const float* chproj_b   = (const float*)d_in[24];
    const float* spatio_w   = (const float*)d_in[25];
    const float* spatio_b   = (const float*)d_in[26];
    const float* up1d_w     = (const float*)d_in[27];
    const float* up1d_b     = (const float*)d_in[28];
    MtaParams mp;
    for (int i = 0; i < 3; ++i) {
        mp.w[i]  = (const float*)d_in[29 + 4 * i];
        mp.b[i]  = (const float*)d_in[30 + 4 * i];
        mp.g[i]  = (const float*)d_in[31 + 4 * i];
        mp.be[i] = (const float*)d_in[32 + 4 * i];
    }
    const float* mta_proj_w = (const float*)d_in[41];
    const float* mta_proj_b = (const float*)d_in[42];
    const float* bn_g       = (const float*)d_in[43];
    const float* bn_b       = (const float*)d_in[44];
    const float* mga_up_w   = (const float*)d_in[45];
    const float* mga_up_b   = (const float*)d_in[46];
    const float* c0_w = (const float*)d_in[47]; const float* c0_b = (const float*)d_in[48];
    const float* c1_w = (const float*)d_in[49]; const float* c1_b = (const float*)d_in[50];
    const float* c2_w = (const float*)d_in[51]; const float* c2_b = (const float*)d_in[52];
    const float* down_w = (const float*)d_in[53]; const float* down_b = (const float*)d_in[54];
    const float* t_w = (const float*)d_in[55]; const float* t_b = (const float*)d_in[56];
    const float* bn1_g = (const float*)d_in[57]; const float* bn1_b = (const float*)d_in[58];
    const float* s_w = (const float*)d_in[59]; const float* s_b = (const float*)d_in[60];
    const float* bn2_g = (const float*)d_in[61]; const float* bn2_b = (const float*)d_in[62];
    const float* out_w = (const float*)d_in[63]; const float* out_b = (const float*)d_in[64];

    // workspace layout
    float* ws = (float*)d_ws;
    const size_t BIG = (size_t)BATCH * QKVC * SEQ;        // 50,331,648 floats
    float* bufA   = ws;                                   // qkv_pre, later vb/av/y
    float* bufQKV = ws + BIG;                             // qkv, later mga_out
    float* small  = ws + 2 * BIG;
    float* mu3    = small; small += (size_t)3 * BATCH * SEQ;
    float* covbuf = small; small += (size_t)BATCH * 3 * 96 * 96;
    float* wgt    = small; small += (size_t)BATCH * DIMC;
    float* xs     = small; small += (size_t)BATCH * SEQ;
    float* ascale = small; small += 64;
    float* invq   = small; small += (size_t)BATCH * DIMC;
    float* invk   = small; small += (size_t)BATCH * DIMC;
    float* logits = small; small += (size_t)BATCH * HEADS * 256;
    float* out2   = small; small += (size_t)BATCH * 5 * (SEQ - 1);

    const size_t ACT = (size_t)BATCH * DIMC * SEQ;        // 16,777,216
    float* vb   = bufA;
    float* av   = bufA + ACT;
    float* ybuf = bufA + 2 * ACT;

    // 1) qkv = W_qkv (768x256) * x  (per batch)
    gemm_wmma_f32<4><<<dim3(32, 12, BATCH), 256, 0, stream>>>(
        qkv_w, 256, 0, x, SEQ, (long)DIMC * SEQ,
        bufA, SEQ, (long)QKVC * SEQ, nullptr, nullptr, QKVC, SEQ, DIMC);
    // 2) depthwise 3x3 (center column only since W=1)
    dwconv_kernel<<<(unsigned)((BIG + 255) / 256), 256, 0, stream>>>(bufA, qkv_dw_w, bufQKV);
    // 3) q,k row norms
    rownorm_kernel<<<2 * BATCH * DIMC, 256, 0, stream>>>(bufQKV, invq, invk);
    // 4) chunk means of v
    chunk_mean_kernel<<<(3 * BATCH * SEQ) / 256, 256, 0, stream>>>(bufQKV, mu3);
    // 5) covariance Grams (WMMA)
    gram_wmma_kernel<<<dim3(5, BATCH * 3), 256, 0, stream>>>(bufQKV, mu3, covbuf);
    // 6) stat + MLP -> channel weights
    cov_stat_mlp_kernel<<<BATCH * 3, 96, 0, stream>>>(covbuf, cp, wgt);
    // 7) chproj GEMM with per-channel weighting fused into B rows
    gemm_wmma_f32<4><<<dim3(32, 4, BATCH), 256, 0, stream>>>(
        chproj_w, 256, 0, bufQKV + (size_t)2 * DIMC * SEQ, SEQ, (long)QKVC * SEQ,
        vb, SEQ, (long)DIMC * SEQ, chproj_b, wgt, DIMC, SEQ, DIMC);
    // 8) spatio reduction
    spatio_kernel<<<(BATCH * SEQ) / 256, 256, 0, stream>>>(vb, spatio_w, spatio_b, xs);
    // 9) MTA per-scale conv+LN+ELU+pool
    mta_scale_kernel<<<dim3(BATCH, 3), 256, 0, stream>>>(xs, up1d_w, up1d_b, mp, ascale);
    // 10) MTA apply (in-place on vb)
    mta_apply_kernel<<<(unsigned)(ACT / 256), 256, 0, stream>>>(
        vb, xs, ascale, up1d_w, up1d_b, mta_proj_w, mta_proj_b);
    // 11) normalized q.k^T (WMMA)
    attn_qk_kernel<<<32, 256, 0, stream>>>(bufQKV, invq, invk, logits);
    // 12) softmax with temperature
    softmax_kernel<<<16, 256, 0, stream>>>(logits, temperature);
    // 13) attn * v (WMMA, 256 batched 16x4096x16 GEMMs)
    gemm_wmma_f32<1><<<dim3(32, 1, BATCH * HEADS), 256, 0, stream>>>(
        logits, 16, 256, vb, SEQ, (long)16 * SEQ,
        av, SEQ, (long)16 * SEQ, nullptr, nullptr, 16, SEQ, 16);
    // 14) output projection
    gemm_wmma_f32<4><<<dim3(32, 4, BATCH), 256, 0, stream>>>(
        proj_w, 256, 0, av, SEQ, (long)DIMC * SEQ,
        ybuf, SEQ, (long)DIMC * SEQ, nullptr, nullptr, DIMC, SEQ, DIMC);
    // 15) fused MGA block
    mga_kernel<<<(unsigned)(ACT / 256), 256, 0, stream>>>(
        ybuf, bufQKV, bn_g, bn_b, mga_up_w, mga_up_b,
        c0_w, c0_b, c1_w, c1_b, c2_w, c2_b, down_w, down_b);
    // 16) stconv
    stconv_kernel<<<dim3((SEQ - 1 + 255) / 256, BATCH), 256, 0, stream>>>(
        bufQKV, t_w, t_b, bn1_g, bn1_b, s_w, s_b, bn2_g, bn2_b, out2);
    // 17) pool + final linear
    final_kernel<<<BATCH, 256, 0, stream>>>(out2, out_w, out_b, (float*)d_out);
}